// SmoothRankAP_48155173323608
// MI455X (gfx1250) — compile-verified
//
#include <hip/hip_runtime.h>

#define BN 512

#if defined(__has_builtin)
#if __has_builtin(__builtin_amdgcn_tanhf)
#define USE_TANH 1
#endif
#endif

#ifdef USE_TANH
// sigmoid((sk-sj)/tau) = 0.5*tanh((sk-sj)/(2*tau)) + 0.5 ; pre-scale by 0.5/tau
#define SCALE 50.0f
#else
// sigmoid((sk-sj)/tau) = rcp(1 + exp2((sj-sk)*log2e/tau)) ; pre-scale by log2e/tau
#define SCALE 144.2695040888963407f
#endif

typedef float v2f __attribute__((ext_vector_type(2)));
typedef float v4f __attribute__((ext_vector_type(4)));
typedef float v8f __attribute__((ext_vector_type(8)));

__device__ __forceinline__ float sig_of(float sk, float sj) {
#ifdef USE_TANH
    return __builtin_fmaf(0.5f, __builtin_amdgcn_tanhf(sk - sj), 0.5f);
#else
    return __builtin_amdgcn_rcpf(1.0f + __builtin_amdgcn_exp2f(sj - sk));
#endif
}

// Grid: (BN, 4). Block: 256 threads = 8 waves. Wave owns 16 rows j of query b.
// 8-k superstep: low half-wave computes sigs for k0..k0+3, high half for
// k0+4..k0+7 (one ds_load_b128 + one global_load_b128 each). Two WMMAs per
// superstep (B = ones, so K-slot -> physical-k mapping is arbitrary) reduce
// the unweighted rank sums on the matrix pipe; VALU FMAs do the
// target-weighted sums.
__global__ __launch_bounds__(256) void srap_main_kernel(
    const float* __restrict__ scores,
    const float* __restrict__ target,
    float* __restrict__ ws)
{
    const int b     = blockIdx.x;
    const int chunk = blockIdx.y;        // 0..3
    const int tid   = threadIdx.x;       // 0..255
    const int wave  = tid >> 5;          // 0..7
    const int lane  = tid & 31;
    const int half  = lane >> 4;         // 0 or 1 (k-subset selector)
    const int l16   = lane & 15;

    __shared__ __align__(16) float ssc[BN];   // scaled scores row of b
    __shared__ __align__(16) float ldsA[8][16];

    for (int i = tid; i < BN; i += 256)
        ssc[i] = scores[(size_t)b * BN + i] * SCALE;
    __syncthreads();

    const int jbase = (chunk * 8 + wave) * 16;
    const int j     = jbase + l16;

    const float sj = ssc[j];
    const float* trow = target + (size_t)j * BN;

    v8f acc0 = {}, acc1 = {};     // two independent WMMA accumulator chains
    v2f bones; bones.x = 1.0f; bones.y = 1.0f;
    float pos0 = 0.0f, pos1 = 0.0f;

    for (int k0 = 0; k0 < BN; k0 += 8) {
        const int kk = k0 + 4 * half;

        v4f sk = *(const v4f*)(ssc + kk);      // broadcast within half-wave
        v4f tv = *(const v4f*)(trow + kk);     // target row j, streaming
        __builtin_prefetch(trow + kk + 128, 0, 3);

        float s0 = sig_of(sk.x, sj);
        float s1 = sig_of(sk.y, sj);
        float s2 = sig_of(sk.z, sj);
        float s3 = sig_of(sk.w, sj);

        pos0 = __builtin_fmaf(s0, tv.x, pos0);
        pos1 = __builtin_fmaf(s1, tv.y, pos1);
        pos0 = __builtin_fmaf(s2, tv.z, pos0);
        pos1 = __builtin_fmaf(s3, tv.w, pos1);

        v2f a0; a0.x = s0; a0.y = s1;          // WMMA #1: k0,k0+1 | k0+4,k0+5
        v2f a1; a1.x = s2; a1.y = s3;          // WMMA #2: k0+2,k0+3 | k0+6,k0+7
        acc0 = __builtin_amdgcn_wmma_f32_16x16x4_f32(
                   false, a0, false, bones, (short)0, acc0, false, false);
        acc1 = __builtin_amdgcn_wmma_f32_16x16x4_f32(
                   false, a1, false, bones, (short)0, acc1, false, false);
    }

    v8f acc = acc0 + acc1;
    float possum = pos0 + pos1;
    // Combine half-wave partials of the weighted sum: P_full[j].
    possum += __shfl_xor(possum, 16, 32);

    // D columns identical (ones-B): lane 0 holds A_full[jbase+0..7],
    // lane 16 holds A_full[jbase+8..15]. Stage to LDS for per-j pickup.
    if (l16 == 0) {
        #pragma unroll
        for (int v = 0; v < 8; ++v)
            ldsA[wave][half * 8 + v] = acc[v];
    }
    __syncthreads();

    // Diagonal identities: sig[j,j] = 0.5 exactly, target[j,j] = 1, so
    //   sim_all = (A_full - 0.5) + 1 = A_full + 0.5
    //   sim_pos = (P_full - 0.5 + target[b,j]) * target[b,j]
    float q = 0.0f;
    if (half == 0) {
        float Afull  = ldsA[wave][l16];
        float tbj    = target[(size_t)b * BN + j];
        float simpos = (possum - 0.5f + tbj) * tbj;
        q = simpos * __builtin_amdgcn_rcpf(Afull + 0.5f);
    }
    q += __shfl_xor(q, 1, 32);
    q += __shfl_xor(q, 2, 32);
    q += __shfl_xor(q, 4, 32);
    q += __shfl_xor(q, 8, 32);
    q += __shfl_xor(q, 16, 32);

    if (lane == 0)
        ws[((size_t)b * 4 + chunk) * 8 + wave] = q;   // 32 partials per b
}

// One block, 512 threads: thread t finalizes ap[b=t], then reduce to loss.
__global__ __launch_bounds__(512) void srap_final_kernel(
    const float* __restrict__ target,
    const float* __restrict__ ws,
    float* __restrict__ out)
{
    __shared__ float red[512];
    const int b = threadIdx.x;

    float s = 0.0f;
    #pragma unroll 8
    for (int i = 0; i < 32; ++i) s += ws[(size_t)b * 32 + i];

    float tsum = 0.0f;
    const float* trow = target + (size_t)b * BN;
    for (int k = 0; k < BN; ++k) tsum += trow[k];

    red[b] = s / tsum;            // ap[b]
    __syncthreads();

    for (int off = 256; off > 0; off >>= 1) {
        if (b < off) red[b] += red[b + off];
        __syncthreads();
    }
    if (b == 0) out[0] = 1.0f - red[0] * (1.0f / (float)BN);
}

extern "C" void kernel_launch(void* const* d_in, const int* in_sizes, int n_in,
                              void* d_out, int out_size, void* d_ws, size_t ws_size,
                              hipStream_t stream) {
    const float* scores = (const float*)d_in[0];
    const float* target = (const float*)d_in[1];
    float* ws  = (float*)d_ws;     // 512*32 floats = 64 KB of scratch
    float* out = (float*)d_out;

    srap_main_kernel<<<dim3(BN, 4), 256, 0, stream>>>(scores, target, ws);
    srap_final_kernel<<<1, 512, 0, stream>>>(target, ws, out);
}